// JACOBIConv_17506286699043
// MI455X (gfx1250) — compile-verified
//
#include <hip/hip_runtime.h>
#include <math.h>

// ---------------- problem constants ----------------
#define NN     100000
#define EE     1250000
#define DD     64
#define KK     3
#define CATD   ((KK + 1) * DD)   // 256

typedef __attribute__((ext_vector_type(2))) float    v2f;
typedef __attribute__((ext_vector_type(8))) float    v8f;
typedef __attribute__((ext_vector_type(4))) unsigned u32x4;
typedef __attribute__((ext_vector_type(4))) int      i32x4;
typedef __attribute__((ext_vector_type(8))) int      i32x8;

// ---------------- utility kernels ----------------
__global__ void zero_f32(float* __restrict__ p, unsigned n) {
    unsigned i = blockIdx.x * blockDim.x + threadIdx.x;
    if (i < n) p[i] = 0.0f;
}

__global__ void deg_count(const int* __restrict__ row, float* __restrict__ deg, int e_cnt) {
    int e = blockIdx.x * blockDim.x + threadIdx.x;
    if (e < e_cnt) atomicAdd(&deg[row[e]], 1.0f);
}

__global__ void deg_to_dis(float* __restrict__ deg, int n) {
    int i = blockIdx.x * blockDim.x + threadIdx.x;
    if (i < n) {
        float d = deg[i];
        if (d < 0.5f) d += 1.0f;
        deg[i] = rsqrtf(d);   // deg^-0.5
    }
}

__global__ void edge_val(const int* __restrict__ row, const int* __restrict__ col,
                         const float* __restrict__ w, const float* __restrict__ dis,
                         float* __restrict__ val, int e_cnt) {
    int e = blockIdx.x * blockDim.x + threadIdx.x;
    if (e < e_cnt) val[e] = dis[row[e]] * w[e] * dis[col[e]];
}

// copy x into column block 0 of xcat [N, 256]
__global__ void copy_x(const float* __restrict__ x, float* __restrict__ xcat, unsigned n) {
    unsigned i = blockIdx.x * blockDim.x + threadIdx.x;   // over N*64
    if (i < n) {
        unsigned node = i >> 6, f = i & 63;
        xcat[(size_t)node * CATD + f] = x[i];
    }
}

// SpMM: Ax[row] += val * h[col]; h = xcat column block colofs, row stride 256.
// One wave per edge (wave32), 2 features per lane -> row/col/val wave-uniform loads.
__global__ void spmm_scatter(const int* __restrict__ row, const int* __restrict__ col,
                             const float* __restrict__ val, const float* __restrict__ xcat,
                             int colofs, float* __restrict__ Ax, unsigned total) {
    unsigned idx = blockIdx.x * blockDim.x + threadIdx.x;   // E*32 threads
    if (idx >= total) return;
    unsigned e = idx >> 5;
    unsigned f = (idx & 31) * 2;
    int r = row[e], c = col[e];
    float v = val[e];
    const float* hp = xcat + (size_t)c * CATD + colofs + f;
    float2 h2 = *(const float2*)hp;   // 8B aligned
    atomicAdd(&Ax[(size_t)r * DD + f],     v * h2.x);
    atomicAdd(&Ax[(size_t)r * DD + f + 1], v * h2.y);
}

// L == 1 recurrence: nxt = alpha0 * (coef1 * x0 + coef2 * Ax)
__global__ void combine1(const float* __restrict__ Ax, const float* __restrict__ alpha_params,
                         float* __restrict__ xcat, unsigned n) {
    unsigned i = blockIdx.x * blockDim.x + threadIdx.x;   // over N*64
    if (i >= n) return;
    const float a = 1.0f, b = 1.0f, l = -1.0f, r = 1.0f;
    float coef1 = (a - b) * 0.5f - (a + b + 2.0f) * 0.5f * ((l + r) / (r - l));
    float coef2 = (a + b + 2.0f) / (r - l);
    float alpha0 = tanhf(alpha_params[0]);
    unsigned node = i >> 6, f = i & 63;
    float x0 = xcat[(size_t)node * CATD + f];
    xcat[(size_t)node * CATD + DD + f] = alpha0 * (coef1 * x0 + coef2 * Ax[i]);
}

// L >= 2 recurrence
__global__ void combineL(const float* __restrict__ Ax, const float* __restrict__ alpha_params,
                         float* __restrict__ xcat, unsigned n, int Lp) {
    unsigned i = blockIdx.x * blockDim.x + threadIdx.x;   // over N*64
    if (i >= n) return;
    const float a = 1.0f, b = 1.0f, l = -1.0f, r = 1.0f;
    float L = (float)Lp;
    float aL1 = tanhf(alpha_params[Lp - 1]);
    float aL2 = tanhf(alpha_params[Lp - 2]);
    float coef_l   = 2.0f * L * (L + a + b) * (2.0f * L - 2.0f + a + b);
    float c_lm1_1  = (2.0f * L + a + b - 1.0f) * (2.0f * L + a + b) * (2.0f * L + a + b - 2.0f);
    float c_lm1_2  = (2.0f * L + a + b - 1.0f) * (a * a - b * b);
    float c_lm2    = 2.0f * (L - 1.0f + a) * (L - 1.0f + b) * (2.0f * L + a + b);
    float tmp1   = aL1 * (c_lm1_1 / coef_l);
    float tmp2   = aL1 * (c_lm1_2 / coef_l);
    float tmp3   = aL1 * aL2 * (c_lm2 / coef_l);
    float tmp1_2 = tmp1 * (2.0f / (r - l));
    float tmp2_2 = tmp1 * ((r + l) / (r - l)) + tmp2;
    unsigned node = i >> 6, f = i & 63;
    const float* xr = xcat + (size_t)node * CATD;
    float xm1 = xr[(Lp - 1) * DD + f];
    float xm2 = xr[(Lp - 2) * DD + f];
    xcat[(size_t)node * CATD + Lp * DD + f] = tmp1_2 * Ax[i] - tmp2_2 * xm1 - tmp3 * xm2;
}

// ---------------- final GEMM: out[N,64] = xcat[N,256] @ W[256,64] + b ----------------
// f32 WMMA 16x16x4. One 16-row tile per wave; each wave produces all 4 16-wide
// column slices (reusing the A fragment).
// W is DMA'd into LDS by the Tensor Data Mover (one wave issues the descriptor,
// s_wait_tensorcnt 0), then transposed in LDS so each B fragment is a
// contiguous 8B pair -> single ds_load_b64, no register shuffles.
__global__ void __launch_bounds__(256)
lin_wmma(const float* __restrict__ W_g,
         const float* __restrict__ xcat,
         const float* __restrict__ bias, float* __restrict__ out, int nrows) {
    __shared__ float smem[2 * CATD * DD];     // [0,16K): raw W (k-major); [16K,32K): transposed
    float* sW  = smem;                        // W[k][n],   k=0..255, n=0..63
    float* sWt = smem + CATD * DD;            // Wt[n][k],  n=0..63,  k=0..255

    int t = threadIdx.x;

#if __has_builtin(__builtin_amdgcn_tensor_load_to_lds) && __has_builtin(__builtin_amdgcn_s_wait_tensorcnt)
    if (t < 32) {                             // wave 0 issues the TDM descriptor
        unsigned lds_off = (unsigned)(size_t)&smem[0];      // low 32b of flat LDS addr == LDS offset
        size_t ga = (size_t)W_g;
        u32x4 g0;
        g0.x = 1u;                                          // count=1, user mode, no gather
        g0.y = lds_off;                                     // lds_addr
        g0.z = (unsigned)(ga & 0xFFFFFFFFu);                // global_addr[31:0]
        g0.w = (unsigned)((ga >> 32) & 0x01FFFFFFu) | (2u << 30);  // addr[56:32] | type=2
        i32x8 g1;
        g1[0] = 0x00020000;                                 // data_size=2 (4B), mask=0
        g1[1] = (int)(((CATD * DD) & 0xFFFF) << 16);        // tensor_dim0 lo -> bits63:48
        g1[2] = 0x00010000;                                 // tensor_dim0 hi=0 | tensor_dim1=1
        g1[3] = (int)(((CATD * DD) & 0xFFFF) << 16);        // tile_dim0 = 16384
        g1[4] = 1;                                          // tile_dim1=1, tile_dim2=0
        g1[5] = CATD * DD;                                  // tensor_dim0_stride
        g1[6] = 0;
        g1[7] = 0;
        i32x4 gz4 = {0, 0, 0, 0};
        i32x8 gz8 = {0, 0, 0, 0, 0, 0, 0, 0};
        // 6-arg variant (clang-23 / therock-10.0 headers)
        __builtin_amdgcn_tensor_load_to_lds(g0, g1, gz4, gz4, gz8, 0);
        __builtin_amdgcn_s_wait_tensorcnt((short)0);
    }
    __syncthreads();
#else
    for (int i = t * 4; i < CATD * DD; i += 256 * 4)
        *(float4*)&sW[i] = *(const float4*)&W_g[i];
    __syncthreads();
#endif

    // transpose inside LDS: sWt[n][k] = sW[k][n]
    for (int i = t; i < CATD * DD; i += 256) {
        int k = i >> 6, n = i & 63;
        sWt[n * CATD + k] = sW[i];
    }
    __syncthreads();

    int wave = t >> 5;
    int lane = t & 31;
    int tile = blockIdx.x * 8 + wave;          // 16-row tile
    if (tile * 16 >= nrows) return;            // wave-uniform: EXEC stays all-ones
    int row0 = tile * 16;

    int mn = lane & 15;                         // M for A-frag, N for B-frag
    int kb = (lane >> 4) * 2;                   // K sub-offset: 0 or 2

    v8f acc0 = {}, acc1 = {}, acc2 = {}, acc3 = {};
    const float* arow = xcat + (size_t)(row0 + mn) * CATD + kb;
    const float* b0p = &sWt[( 0 + mn) * CATD + kb];
    const float* b1p = &sWt[(16 + mn) * CATD + kb];
    const float* b2p = &sWt[(32 + mn) * CATD + kb];
    const float* b3p = &sWt[(48 + mn) * CATD + kb];

    for (int k = 0; k < CATD; k += 4) {
        v2f a  = *(const v2f*)&arow[k];     // global_load_b64
        v2f b0 = *(const v2f*)&b0p[k];      // ds_load_b64 (contiguous pair)
        v2f b1 = *(const v2f*)&b1p[k];
        v2f b2 = *(const v2f*)&b2p[k];
        v2f b3 = *(const v2f*)&b3p[k];
        acc0 = __builtin_amdgcn_wmma_f32_16x16x4_f32(false, a, false, b0, (short)0, acc0, false, false);
        acc1 = __builtin_amdgcn_wmma_f32_16x16x4_f32(false, a, false, b1, (short)0, acc1, false, false);
        acc2 = __builtin_amdgcn_wmma_f32_16x16x4_f32(false, a, false, b2, (short)0, acc2, false, false);
        acc3 = __builtin_amdgcn_wmma_f32_16x16x4_f32(false, a, false, b3, (short)0, acc3, false, false);
    }

    // C/D layout: VGPR v -> M = v + 8*(lane>=16); N = lane&15
    int mhi = (lane >> 4) * 8;
    #pragma unroll
    for (int v = 0; v < 8; ++v) {
        int r = row0 + v + mhi;
        float* orow = out + (size_t)r * DD;
        orow[ 0 + mn] = acc0[v] + bias[ 0 + mn];
        orow[16 + mn] = acc1[v] + bias[16 + mn];
        orow[32 + mn] = acc2[v] + bias[32 + mn];
        orow[48 + mn] = acc3[v] + bias[48 + mn];
    }
}

// ---------------- host launch ----------------
extern "C" void kernel_launch(void* const* d_in, const int* in_sizes, int n_in,
                              void* d_out, int out_size, void* d_ws, size_t ws_size,
                              hipStream_t stream) {
    const float* x       = (const float*)d_in[0];
    const int*   eidx    = (const int*)d_in[1];     // [2,E]: row then col
    const float* eweight = (const float*)d_in[2];
    const float* alphas  = (const float*)d_in[3];
    const float* lin_w   = (const float*)d_in[4];
    const float* lin_b   = (const float*)d_in[5];
    float* out = (float*)d_out;

    const int* row = eidx;
    const int* col = eidx + EE;

    // workspace carve-up
    float* xcat = (float*)d_ws;                         // N*256
    float* Ax   = xcat + (size_t)NN * CATD;             // N*64
    float* deg  = Ax + (size_t)NN * DD;                 // N (deg -> dis in place)
    float* val  = deg + NN;                             // E

    const int T = 256;
    unsigned nfeat = (unsigned)NN * DD;                 // 6.4M

    // 1) degree -> dis
    zero_f32<<<(NN + T - 1) / T, T, 0, stream>>>(deg, NN);
    deg_count<<<(EE + T - 1) / T, T, 0, stream>>>(row, deg, EE);
    deg_to_dis<<<(NN + T - 1) / T, T, 0, stream>>>(deg, NN);

    // 2) edge values
    edge_val<<<(EE + T - 1) / T, T, 0, stream>>>(row, col, eweight, deg, val, EE);

    // 3) xs[0] = x
    copy_x<<<(nfeat + T - 1) / T, T, 0, stream>>>(x, xcat, nfeat);

    // 4) Jacobi recursion
    unsigned spmm_threads = (unsigned)EE * 32;          // 40M
    for (int L = 1; L <= KK; ++L) {
        zero_f32<<<(nfeat + T - 1) / T, T, 0, stream>>>(Ax, nfeat);
        spmm_scatter<<<(spmm_threads + T - 1) / T, T, 0, stream>>>(
            row, col, val, xcat, (L - 1) * DD, Ax, spmm_threads);
        if (L == 1)
            combine1<<<(nfeat + T - 1) / T, T, 0, stream>>>(Ax, alphas, xcat, nfeat);
        else
            combineL<<<(nfeat + T - 1) / T, T, 0, stream>>>(Ax, alphas, xcat, nfeat, L);
    }

    // 5) final linear layer via f32 WMMA: 6250 row tiles, 8 waves/block
    int tiles = NN / 16;                                // 6250
    int blocks = (tiles + 7) / 8;                       // 782
    lin_wmma<<<blocks, 256, 0, stream>>>(lin_w, xcat, lin_b, out, NN);
}